// FixedDilatedCNN_27779848470899
// MI455X (gfx1250) — compile-verified
//
#include <hip/hip_runtime.h>

// ---------------------------------------------------------------------------
// FixedDilatedCNN on MI455X (gfx1250)
//
// out[n,i,5(s-2)+v] = x[n,i+f,:] . (W * rev(x[n,i-b,:]))  (+ GU term, v==0)
// Reformulated: P[j,k] = x[j,:] . A0[k,:] with A0[k,c] = w[c]*x[k,3-c]
//  -> a K=4 f32 GEMM == V_WMMA_F32_16X16X4_F32 tiles along the band
//     j-k in [6,2050]. Each 16x16 tile covers 6 s-values for an 8-row
//     output block; extraction writes contiguous 5-float groups.
// Store-bound problem (335 MB out); input window staged in LDS (33 KB)
// via global_load_async_to_lds_b128 (ASYNCcnt).
// ---------------------------------------------------------------------------

typedef float v2f __attribute__((ext_vector_type(2)));
typedef float v8f __attribute__((ext_vector_type(8)));
typedef int   v4i __attribute__((ext_vector_type(4)));

#define BATCH   8
#define LEN     2048
#define NF      5110       // 5 * 1022 features
#define ROWS    8          // output rows (i) per workgroup
#define WIN     2080       // staged window length (vec4 elems), zero-padded
#define WOFF    1032       // window start = i0 - WOFF
#define NCHUNK  171        // ceil(1022 s-values / 6 per tile)

__global__ __launch_bounds__(128)
void fdcnn_wmma_kernel(const float4* __restrict__ x4, float* __restrict__ out)
{
    __shared__ float4 xw[WIN];        // zero-padded x window, 33.3 KB
    __shared__ float  slab[4][256];   // one 16x16 D-tile per wave

    const int tid  = threadIdx.x;
    const int wave = tid >> 5;
    const int lane = tid & 31;
    const int n    = blockIdx.x >> 8;            // 256 i-blocks per batch
    const int i0   = (blockIdx.x & 255) * ROWS;  // first output row
    const int wbase = i0 - WOFF;                 // global index of xw[0]

    // ---- Phase 0: stage zero-padded x[n, wbase : wbase+WIN) into LDS ----
#if __has_builtin(__builtin_amdgcn_global_load_async_to_lds_b128) && \
    __has_builtin(__builtin_amdgcn_s_wait_asynccnt)
    for (int idx = tid; idx < WIN; idx += 128) {
        float4 z = {0.f, 0.f, 0.f, 0.f};
        xw[idx] = z;                              // zero-fill (padding)
    }
    __syncthreads();
    for (int idx = tid; idx < WIN; idx += 128) {
        const int g = wbase + idx;
        if (g >= 0 && g < LEN) {
            __builtin_amdgcn_global_load_async_to_lds_b128(
                (__attribute__((address_space(1))) v4i*)(x4 + (size_t)n * LEN + g),
                (__attribute__((address_space(3))) v4i*)&xw[idx],
                0, 0);
        }
    }
    __builtin_amdgcn_s_wait_asynccnt(0);
    __syncthreads();
#else
    for (int idx = tid; idx < WIN; idx += 128) {
        const int g = wbase + idx;
        float4 v = {0.f, 0.f, 0.f, 0.f};
        if (g >= 0 && g < LEN) v = x4[(size_t)n * LEN + g];
        xw[idx] = v;
    }
    __syncthreads();
#endif

    // ---- Phase 1: one 16x16 WMMA P-tile per 6-s chunk, then extract ----
    float* myslab = slab[wave];
    const int h  = lane >> 4;   // half-wave: selects channel pair (2h,2h+1)
    const int mp = lane & 15;   // row (A) / col (B,D) within tile

    for (int c = wave; c < NCHUNK; c += 4) {
        const int s_base = 2 + 6 * c;
        const int sigma  = s_base + 1;
        const int jl = sigma + WOFF;        // local j0 = (i0+sigma) - wbase
        const int kl = WOFF - sigma - 7;    // local k0 = (i0-sigma-7) - wbase

        // A fragment (16x4): lane<16 -> channels (0,1), lane>=16 -> (2,3)
        v2f a = ((const v2f*)&xw[jl + mp])[h];

        // B fragment (4x16): B[c][p] = w[c] * z[p][3-c], w = {2,3,3,2}
        v2f zl = ((const v2f*)&xw[kl + mp])[0];   // (z0, z1)
        v2f zh = ((const v2f*)&xw[kl + mp])[1];   // (z2, z3)
        v2f b;
        if (h == 0) { b.x = 2.f * zh.y; b.y = 3.f * zh.x; }  // rows K=0,1
        else        { b.x = 3.f * zl.y; b.y = 2.f * zl.x; }  // rows K=2,3

        v8f acc = {};
        acc = __builtin_amdgcn_wmma_f32_16x16x4_f32(
                  false, a, false, b, (short)0, acc, false, false);

        // Spill D to this wave's LDS slab: lane holds col p=mp, rows q+8h
#pragma unroll
        for (int q = 0; q < 8; ++q)
            myslab[(q + 8 * h) * 16 + mp] = acc[q];

        __builtin_amdgcn_wave_barrier();

        // Extract 6s x 8i x 5v = 240 values; lanes 0..29 cover (s_off, v)
        if (lane < 30) {
            const int s_off = lane / 5;
            const int v     = lane - 5 * s_off;
            const int s     = s_base + s_off;
            if (s <= 1023) {
                const int dj = (v == 3) ? 1 : ((v == 4) ? 2 : 0);
                const int dk = (v == 1) ? 1 : ((v == 2) ? 2 : 0);
#pragma unroll
                for (int io = 0; io < ROWS; ++io) {
                    const int i  = i0 + io;
                    const int jj = io + s + 1 + dj;       // j - i0
                    const int kk = io - s - 1 - dk;       // k - i0
                    const int m  = jj - sigma;            // tile row
                    const int p  = kk + sigma + 7;        // tile col
                    float val = myslab[m * 16 + p];
                    if (v == 0) {
                        // GU wobble: + 2*(y2*z3 + y3*z2), main variant only
                        v2f yh  = ((const v2f*)&xw[jj + WOFF])[1];
                        v2f zh2 = ((const v2f*)&xw[kk + WOFF])[1];
                        val += 2.f * (yh.x * zh2.y + yh.y * zh2.x);
                    }
                    out[(size_t)(n * LEN + i) * NF + 5 * (s - 2) + v] = val;
                }
            }
        }
        __builtin_amdgcn_wave_barrier();
    }
}

extern "C" void kernel_launch(void* const* d_in, const int* in_sizes, int n_in,
                              void* d_out, int out_size, void* d_ws, size_t ws_size,
                              hipStream_t stream)
{
    (void)in_sizes; (void)n_in; (void)out_size; (void)d_ws; (void)ws_size;
    const float4* x4 = (const float4*)d_in[0];   // (8, 2048, 4) f32, 16B rows
    float* out = (float*)d_out;                  // (8, 2048, 5110) f32
    // 8 batches * 256 eight-row blocks = 2048 workgroups, 4 waves each
    fdcnn_wmma_kernel<<<dim3(BATCH * (LEN / ROWS)), dim3(128), 0, stream>>>(x4, out);
}